// ModelNew_16776142258799
// MI455X (gfx1250) — compile-verified
//
#include <hip/hip_runtime.h>

// MI455X (gfx1250) — pure HBM-bound reduce+broadcast.
// 256 MB total traffic @ 23.3 TB/s => ~11 us floor. Accumulation rides the
// matrix pipe: V_WMMA_F32_16X16X4_F32 with A = ones gives sum(D) = 16*sum(B),
// layout-independent, fp32 accumulation. Input/output streams are marked
// non-temporal so the 2x128 MB streams don't thrash the 192 MB L2; only the
// 4 KB partials array (real reuse: re-read by every fill block) stays RT/hot.

typedef __attribute__((ext_vector_type(2))) float v2f;
typedef __attribute__((ext_vector_type(4))) float v4f;
typedef __attribute__((ext_vector_type(8))) float v8f;

#define BLOCK 256
#define GRID_REDUCE 1024
#define GRID_FILL 2048

__global__ __launch_bounds__(BLOCK) void ModelNew_reduce_kernel(
    const float* __restrict__ x, float* __restrict__ partials, int n) {
  const int tid = threadIdx.x;
  const int gid = blockIdx.x * BLOCK + tid;
  const int nthreads = GRID_REDUCE * BLOCK;

  const int nvec = n >> 2;                 // float4 elements
  const v4f* __restrict__ xv = (const v4f*)x;

  // WMMA accumulator (16x16 f32 C/D = 8 VGPRs) and all-ones A matrix (16x4 f32
  // = 2 VGPRs; every lane holds 1.0,1.0 => A is all ones).
  v8f acc = {};
  v2f ones;
  ones.x = 1.0f;
  ones.y = 1.0f;

  // Uniform trip count across the whole grid => EXEC stays all-1s through the
  // WMMA loop (ISA requirement for V_WMMA_*).
  const int full_iters = nvec / nthreads;
  int i = gid;
  for (int it = 0; it < full_iters; ++it) {
    // Prefetch a few grid-strides ahead (global_prefetch_b8) to keep the HBM
    // read pipeline deep while the XDL pipe chews the accumulation.
    __builtin_prefetch(&xv[i + 4 * nthreads], 0, 1);
    v4f d = __builtin_nontemporal_load(&xv[i]);  // global_load_b128 th:NT
    v2f b0; b0.x = d.x; b0.y = d.y;              // B = 4x16 f32 tile (2 VGPRs)
    v2f b1; b1.x = d.z; b1.y = d.w;
    // D = A(ones) * B + C  => each B element counted 16x across the D tile.
    acc = __builtin_amdgcn_wmma_f32_16x16x4_f32(
        false, ones, false, b0, (short)0, acc, false, false);
    acc = __builtin_amdgcn_wmma_f32_16x16x4_f32(
        false, ones, false, b1, (short)0, acc, false, false);
    i += nthreads;
  }

  // Non-uniform remainder: plain VALU adds (no WMMA under divergence).
  float extra = 0.0f;
  if (i < nvec) {
    v4f d = __builtin_nontemporal_load(&xv[i]);
    extra = (d.x + d.y) + (d.z + d.w);
  }
  // Scalar tail for n not divisible by 4.
  for (int j = (nvec << 2) + gid; j < n; j += nthreads) extra += x[j];

  // sum(D tile) = 16 * accumulated sum(B). Per-lane contribution:
  float lane = (((acc[0] + acc[1]) + (acc[2] + acc[3])) +
                ((acc[4] + acc[5]) + (acc[6] + acc[7]))) * 0.0625f +
               extra;

  // Deterministic block tree reduction in LDS.
  __shared__ float smem[BLOCK];
  smem[tid] = lane;
  __syncthreads();
  for (int s = BLOCK / 2; s > 0; s >>= 1) {
    if (tid < s) smem[tid] += smem[tid + s];
    __syncthreads();
  }
  if (tid == 0) partials[blockIdx.x] = smem[0];  // RT store: keep L2-hot
}

__global__ __launch_bounds__(BLOCK) void ModelNew_fill_kernel(
    const float* __restrict__ partials, float* __restrict__ out, int n) {
  // Every block re-reduces the 1024 partials (4 KB, L2-hot) with a fixed tree
  // => bit-identical total in every block, no atomics, fully deterministic.
  __shared__ float smem[BLOCK];
  const int tid = threadIdx.x;
  float v = 0.0f;
  for (int i = tid; i < GRID_REDUCE; i += BLOCK) v += partials[i];
  smem[tid] = v;
  __syncthreads();
  for (int s = BLOCK / 2; s > 0; s >>= 1) {
    if (tid < s) smem[tid] += smem[tid + s];
    __syncthreads();
  }
  const float total = smem[0];

  v4f t4;
  t4.x = total; t4.y = total; t4.z = total; t4.w = total;
  const int nvec = n >> 2;
  v4f* __restrict__ ov = (v4f*)out;
  const int stride = GRID_FILL * BLOCK;
  for (int i = blockIdx.x * BLOCK + tid; i < nvec; i += stride)
    __builtin_nontemporal_store(t4, &ov[i]);     // global_store_b128 th:NT
  for (int j = (nvec << 2) + blockIdx.x * BLOCK + tid; j < n; j += stride)
    out[j] = total;
}

extern "C" void kernel_launch(void* const* d_in, const int* in_sizes, int n_in,
                              void* d_out, int out_size, void* d_ws, size_t ws_size,
                              hipStream_t stream) {
  const float* x = (const float*)d_in[0];
  float* out = (float*)d_out;
  float* partials = (float*)d_ws;          // GRID_REDUCE floats = 4 KB scratch
  const int n = in_sizes[0];

  ModelNew_reduce_kernel<<<GRID_REDUCE, BLOCK, 0, stream>>>(x, partials, n);
  ModelNew_fill_kernel<<<GRID_FILL, BLOCK, 0, stream>>>(partials, out, out_size);
}